// GraphDecoderHomo_76201309765988
// MI455X (gfx1250) — compile-verified
//
#include <hip/hip_runtime.h>
#include <hip/hip_bf16.h>

typedef __attribute__((ext_vector_type(2))) float v2f;
typedef __attribute__((ext_vector_type(8))) float v8f;

// ---------------------------------------------------------------------------
// zero fill
// ---------------------------------------------------------------------------
__global__ void zero_f32(float* __restrict__ p, int n) {
    int i = blockIdx.x * blockDim.x + threadIdx.x;
    if (i < n) p[i] = 0.0f;
}

// ---------------------------------------------------------------------------
// in-degree (float, matches segment_sum of ones)
// ---------------------------------------------------------------------------
__global__ void degree_k(const int* __restrict__ dst, float* __restrict__ deg, int E) {
    int e = blockIdx.x * blockDim.x + threadIdx.x;
    if (e < E) atomicAdd(&deg[dst[e]], 1.0f);
}

// ---------------------------------------------------------------------------
// edge scatter: agg[dst] += ew * h[src]   (float4 gather -> 4 f32 atomics)
// one thread per (edge, 4-feature chunk)
// ---------------------------------------------------------------------------
__global__ void scatter_k(const float* __restrict__ h, const int* __restrict__ src,
                          const int* __restrict__ dst, const float* __restrict__ ew,
                          float* __restrict__ agg, int E, int K) {
    int tid = blockIdx.x * blockDim.x + threadIdx.x;
    int chunks = K >> 2;
    int total = E * chunks;
    if (tid >= total) return;
    int e = tid / chunks;
    int c = (tid - e * chunks) << 2;

    int s = src[e];
    int d = dst[e];
    float w = ew[e];

    float4 hv = *(const float4*)(h + (size_t)s * K + c);
    float* ap = agg + (size_t)d * K + c;
    atomicAdd(ap + 0, hv.x * w);
    atomicAdd(ap + 1, hv.y * w);
    atomicAdd(ap + 2, hv.z * w);
    atomicAdd(ap + 3, hv.w * w);
}

// ---------------------------------------------------------------------------
// Fused SAGE dense stage, fp32 WMMA (V_WMMA_F32_16X16X4_F32):
//   out = act( h @ Ws + (agg/max(deg,1)) @ Wn + b )
// One wave (32 lanes) computes a 16(node) x 16(feature) tile.
//
// A-matrix 16x4 f32 layout (ISA 7.12.2): lane = M%16 for lanes 0..15 (K=0,1)
// and lanes 16..31 (K=2,3); VGPR0=first K, VGPR1=second K.
// C/D 16x16 f32: VGPR j -> row j (lanes 0-15), row j+8 (lanes 16-31).
// ---------------------------------------------------------------------------
__global__ void sage_gemm(const float* __restrict__ h, const float* __restrict__ agg,
                          const float* __restrict__ deg,
                          const float* __restrict__ Ws, const float* __restrict__ Wn,
                          const float* __restrict__ bias,
                          float* __restrict__ out,
                          int K, int Fout, int activate) {
    int lane = threadIdx.x;            // 0..31
    int m    = lane & 15;
    int kh   = (lane >> 4) << 1;       // 0 for lanes 0-15, 2 for lanes 16-31
    int row0 = blockIdx.x << 4;
    int col0 = blockIdx.y << 4;

    int node = row0 + m;
    int col  = col0 + m;
    // branch-free column masking (keeps EXEC all-1s for WMMA)
    int   cs    = (col < Fout) ? col  : 0;
    float cmask = (col < Fout) ? 1.0f : 0.0f;

    float invdeg = 1.0f / fmaxf(deg[node], 1.0f);

    const float* hrow = h   + (size_t)node * K;
    const float* arow = agg + (size_t)node * K;

    v8f c = {};
    for (int k = 0; k < K; k += 4) {
        v2f a, b;
        // self path: h @ Ws
        a.x = hrow[k + kh];
        a.y = hrow[k + kh + 1];
        b.x = Ws[(k + kh)     * Fout + cs] * cmask;
        b.y = Ws[(k + kh + 1) * Fout + cs] * cmask;
        c = __builtin_amdgcn_wmma_f32_16x16x4_f32(false, a, false, b,
                                                  (short)0, c, false, false);
        // neighbor path: (agg * invdeg) @ Wn
        a.x = arow[k + kh]     * invdeg;
        a.y = arow[k + kh + 1] * invdeg;
        b.x = Wn[(k + kh)     * Fout + cs] * cmask;
        b.y = Wn[(k + kh + 1) * Fout + cs] * cmask;
        c = __builtin_amdgcn_wmma_f32_16x16x4_f32(false, a, false, b,
                                                  (short)0, c, false, false);
    }

    int half = lane >> 4;
    int cn   = col0 + (lane & 15);
    if (cn < Fout) {
        float bv = bias[cn];
        for (int j = 0; j < 8; ++j) {
            int rm = row0 + j + (half << 3);
            float v = c[j] + bv;
            if (activate) v = tanhf(v);
            out[(size_t)rm * Fout + cn] = v;
        }
    }
}

// ---------------------------------------------------------------------------
// Host-side orchestration
// ---------------------------------------------------------------------------
extern "C" void kernel_launch(void* const* d_in, const int* in_sizes, int n_in,
                              void* d_out, int out_size, void* d_ws, size_t ws_size,
                              hipStream_t stream) {
    const float* b_z  = (const float*)d_in[0];
    const int*   esrc = (const int*)  d_in[1];
    const int*   edst = (const int*)  d_in[2];
    const float* ew   = (const float*)d_in[3];
    const float* Ws0  = (const float*)d_in[4];
    const float* Wn0  = (const float*)d_in[5];
    const float* b0   = (const float*)d_in[6];
    const float* Ws1  = (const float*)d_in[7];
    const float* Wn1  = (const float*)d_in[8];
    const float* b1   = (const float*)d_in[9];
    const float* Ws2  = (const float*)d_in[10];
    const float* Wn2  = (const float*)d_in[11];
    const float* b2   = (const float*)d_in[12];

    const int N = in_sizes[0] / 64;   // 50000 (multiple of 16)
    const int E = in_sizes[1];        // 800000

    // workspace layout (all 16B-aligned)
    float* deg = (float*)d_ws;                 // N
    float* h1  = deg + N;                      // N*32
    float* h2  = h1  + (size_t)N * 32;         // N*16
    float* agg = h2  + (size_t)N * 16;         // N*64 (max, reused each layer)
    float* out = (float*)d_out;                // N*1

    auto cdiv = [](int a, int b) { return (a + b - 1) / b; };

    // in-degree once (identical across layers)
    zero_f32<<<cdiv(N, 256), 256, 0, stream>>>(deg, N);
    degree_k<<<cdiv(E, 256), 256, 0, stream>>>(edst, deg, E);

    struct Layer {
        const float *hin, *Ws, *Wn, *b;
        float* hout;
        int K, F, act;
    } L[3] = {
        { b_z, Ws0, Wn0, b0, h1,  64, 32, 1 },
        { h1,  Ws1, Wn1, b1, h2,  32, 16, 1 },
        { h2,  Ws2, Wn2, b2, out, 16,  1, 0 },
    };

    for (int i = 0; i < 3; ++i) {
        const int K = L[i].K, F = L[i].F;
        zero_f32<<<cdiv(N * K, 256), 256, 0, stream>>>(agg, N * K);
        scatter_k<<<cdiv(E * (K / 4), 256), 256, 0, stream>>>(
            L[i].hin, esrc, edst, ew, agg, E, K);
        dim3 grid(N / 16, cdiv(F, 16));
        sage_gemm<<<grid, 32, 0, stream>>>(
            L[i].hin, agg, deg, L[i].Ws, L[i].Wn, L[i].b, L[i].hout, K, F, L[i].act);
    }
}